// HeteroGraphConvGNN_32865089749543
// MI455X (gfx1250) — compile-verified
//
#include <hip/hip_runtime.h>

// HeteroGraphConvGNN for MI455X (gfx1250, wave32).
// Phase 1: zero agg workspace. Phase 2: scatter-add x[src] -> agg[dst]
// (global_atomic_add_f32, wave-per-edge). Phase 3: fused dual-GraphConv +
// bias + ReLU + Wfc dot using V_WMMA_F32_16X16X4_F32; weights staged in LDS
// via GLOBAL_LOAD_ASYNC_TO_LDS_B128 (ASYNCcnt) with s_wait_asynccnt.

typedef float v2f __attribute__((ext_vector_type(2)));
typedef float v8f __attribute__((ext_vector_type(8)));

#define D_IN 128
#define D_OUT 64
#define LDS_STRIDE 132                    // 128 + 4 pad: bank-conflict-free b64 frags
#define MAT_LDS (D_OUT * LDS_STRIDE)      // floats per staged 64x128 matrix
#define GEMM_BLOCK 256
#define WAVES_PB 8                        // 8 waves * 16 rows = 128 rows / block
#define ASYNC_STAGE 1                     // use global_load_async_to_lds_b128

__global__ void zero_f4(float4* __restrict__ p, long long n4) {
  long long i = (long long)blockIdx.x * blockDim.x + threadIdx.x;
  long long stride = (long long)gridDim.x * blockDim.x;
  float4 z = make_float4(0.f, 0.f, 0.f, 0.f);
  for (; i < n4; i += stride) p[i] = z;
}

// One wave per edge: lane c handles channels c, c+32, c+64, c+96.
__global__ void scatter_add(const float* __restrict__ x,
                            const int* __restrict__ src,
                            const int* __restrict__ dst,
                            float* __restrict__ agg, int nE) {
  int lane = threadIdx.x & 31;
  int wid = (int)((blockIdx.x * blockDim.x + threadIdx.x) >> 5);
  int nwaves = (int)((gridDim.x * blockDim.x) >> 5);
  for (int e = wid; e < nE; e += nwaves) {
    int s = src[e];           // uniform -> scalar load
    int d = dst[e];
    if (e + nwaves < nE)      // pull next gather row toward the WGP
      __builtin_prefetch(x + (long long)src[e + nwaves] * D_IN, 0, 0);
    const float* xr = x + (long long)s * D_IN;
    float* ar = agg + (long long)d * D_IN;
#pragma unroll
    for (int c = 0; c < D_IN; c += 32)
      unsafeAtomicAdd(&ar[c + lane], xr[c + lane]);   // global_atomic_add_f32
  }
}

__global__ __launch_bounds__(GEMM_BLOCK)
void gnn_fused(const float* __restrict__ x0, const float* __restrict__ x1,
               const float* __restrict__ agg0, const float* __restrict__ agg1,
               const float* __restrict__ Wrel0, const float* __restrict__ brel0,
               const float* __restrict__ Wroot0,
               const float* __restrict__ Wrel1, const float* __restrict__ brel1,
               const float* __restrict__ Wroot1,
               const float* __restrict__ Wfc, const float* __restrict__ bfc,
               float* __restrict__ out, int nNodes) {
  extern __shared__ float smem[];   // 4 matrices, 64 x 132 floats each (132 KB)

  // ---- Stage 4 weight matrices [64x128] -> LDS [64x132] (padded stride) ----
  // Async DMA: 16B per lane per issue; no VGPR data registers consumed.
  const float* mats[4] = {Wrel0, Wroot0, Wrel1, Wroot1};
#pragma unroll
  for (int mi = 0; mi < 4; ++mi) {
    const float* srcm = mats[mi];
    for (int c = threadIdx.x; c < (D_OUT * D_IN / 4); c += GEMM_BLOCK) {
      int row = c >> 5;              // 32 float4 chunks per 128-float row
      int col = (c & 31) << 2;
      const float* g = srcm + row * D_IN + col;
#if ASYNC_STAGE
      int ldsByte = (mi * MAT_LDS + row * LDS_STRIDE + col) * 4;
      asm volatile("global_load_async_to_lds_b128 %0, %1, off"
                   :: "v"(ldsByte), "v"(g) : "memory");
#else
      float4 v = *(const float4*)g;
      // row*132 is a multiple of 4 floats -> 16B-aligned LDS store (b128)
      *(float4*)&smem[mi * MAT_LDS + row * LDS_STRIDE + col] = v;
#endif
    }
  }
#if ASYNC_STAGE
  asm volatile("s_wait_asynccnt 0x0" ::: "memory");  // my async writes landed
#endif
  __syncthreads();                                   // everyone's writes visible

  int wave = threadIdx.x >> 5;
  int lane = threadIdx.x & 31;
  int laneRow = lane & 15;           // A: M row / B: N column
  int laneHi = lane >> 4;            // half-wave selects K pair {0,1} vs {2,3}
  int koff = laneHi * 2;
  int m0 = (blockIdx.x * WAVES_PB + wave) * 16;
  if (m0 >= nNodes) return;          // wave-uniform exit (keeps EXEC full for WMMA)
  int mrow = m0 + laneRow;
  if (mrow >= nNodes) mrow = nNodes - 1;   // clamp loads; tail rows not stored

  float psum[8];
#pragma unroll
  for (int j = 0; j < 8; ++j) psum[j] = 0.f;

#pragma unroll
  for (int r = 0; r < 2; ++r) {
    const float* As[2] = {r ? agg1 : agg0, r ? x1 : x0};     // K = 128 + 128
    const float* brel = r ? brel1 : brel0;

    v8f acc[4] = {};   // 4 N-tiles of 16 columns -> full D_OUT = 64

#pragma unroll
    for (int half = 0; half < 2; ++half) {   // half 0: agg@Wrel, half 1: x@Wroot
      const float* A = As[half] + (long long)mrow * D_IN + koff;
      const float* W = &smem[(2 * r + half) * MAT_LDS + laneRow * LDS_STRIDE + koff];
#pragma unroll 4
      for (int ks = 0; ks < D_IN / 4; ++ks) {
        int k0 = ks * 4;
        v2f a = *(const v2f*)(A + k0);       // A frag: 16x4 f32
#pragma unroll
        for (int t = 0; t < 4; ++t) {
          v2f b = *(const v2f*)(W + t * 16 * LDS_STRIDE + k0);  // B frag from LDS
#if __has_builtin(__builtin_amdgcn_wmma_f32_16x16x4_f32)
          acc[t] = __builtin_amdgcn_wmma_f32_16x16x4_f32(
              false, a, false, b, (short)0, acc[t], false, false);
#else
          // compile-only fallback (not layout-correct)
          acc[t][0] += a[0] * b[0] + a[1] * b[1];
#endif
        }
      }
    }

    // bias + ReLU + dot with Wfc segment, all in registers.
    // C layout: lane N = laneRow, VGPR j holds M = j + 8*laneHi.
#pragma unroll
    for (int t = 0; t < 4; ++t) {
      int n = t * 16 + laneRow;
      float wfcv = Wfc[r * D_OUT + n];
      float bb = brel[n];
#pragma unroll
      for (int j = 0; j < 8; ++j) {
        float h = acc[t][j] + bb;
        h = fmaxf(h, 0.f);
        psum[j] = fmaf(wfcv, h, psum[j]);
      }
    }
  }

  // Reduce over the 16 lanes sharing each output row (wave32 halves).
  float bias = bfc[0];
#pragma unroll
  for (int j = 0; j < 8; ++j) {
    float v = psum[j];
    v += __shfl_xor(v, 8, 16);
    v += __shfl_xor(v, 4, 16);
    v += __shfl_xor(v, 2, 16);
    v += __shfl_xor(v, 1, 16);
    if (laneRow == 0) {
      int row = m0 + j + 8 * laneHi;
      if (row < nNodes) out[row] = v + bias;
    }
  }
}

extern "C" void kernel_launch(void* const* d_in, const int* in_sizes, int n_in,
                              void* d_out, int out_size, void* d_ws, size_t ws_size,
                              hipStream_t stream) {
  const float* x0    = (const float*)d_in[0];
  const float* x1    = (const float*)d_in[1];
  const int*   e0    = (const int*)d_in[2];   // [2,E]: row 0 = src, row 1 = dst
  const int*   e1    = (const int*)d_in[3];
  const float* Wrel0 = (const float*)d_in[4];
  const float* brel0 = (const float*)d_in[5];
  const float* Wroot0= (const float*)d_in[6];
  const float* Wrel1 = (const float*)d_in[7];
  const float* brel1 = (const float*)d_in[8];
  const float* Wroot1= (const float*)d_in[9];
  const float* Wfc   = (const float*)d_in[10];
  const float* bfc   = (const float*)d_in[11];
  float* out = (float*)d_out;

  int nNodes = in_sizes[0] / D_IN;
  int nE = in_sizes[2] / 2;

  float* agg0 = (float*)d_ws;
  float* agg1 = agg0 + (long long)nNodes * D_IN;

  long long n4 = (long long)nNodes * D_IN * 2 / 4;
  zero_f4<<<2048, 256, 0, stream>>>((float4*)d_ws, n4);

  scatter_add<<<4096, 256, 0, stream>>>(x0, e0, e0 + nE, agg0, nE);
  scatter_add<<<4096, 256, 0, stream>>>(x1, e1, e1 + nE, agg1, nE);

  int blocks = (nNodes + WAVES_PB * 16 - 1) / (WAVES_PB * 16);
  size_t ldsBytes = 4 * MAT_LDS * sizeof(float);   // 132 KB of 320 KB WGP LDS
  gnn_fused<<<blocks, GEMM_BLOCK, ldsBytes, stream>>>(
      x0, x1, agg0, agg1, Wrel0, brel0, Wroot0, Wrel1, brel1, Wroot1,
      Wfc, bfc, out, nNodes);
}